// MLPSimDirectConstructor_26508538151437
// MI455X (gfx1250) — compile-verified
//
#include <hip/hip_runtime.h>

// MI455X / gfx1250 implementation.
// Structure:
//   1) prep_nv:    nv1/nv2 = tanh(3 * S @ W^T)            [B,N,32] each
//   2) prep_s:     four rank-1 vectors for st/ts blocks
//   3) ss_wmma:    16x16 tiles of tanh(relu(tanh(m - m^T))) via V_WMMA_F32_16X16X4_F32
//   4) tt_wmma:    16x16 tiles of tanh(relu(triu(Tn @ Tn^T)))  (same WMMA path)
//   5) fill_st_ts: broadcast-add blocks, one thread writes one st + one ts element
// Output-bandwidth bound (87 MB @ 23.3 TB/s); WMMA covers the K=32 reductions.

#define LD 32          // latent dim D
#define ALPHA_C 3.0f

typedef float v2f __attribute__((ext_vector_type(2)));
typedef float v8f __attribute__((ext_vector_type(8)));

__device__ __forceinline__ float act_relu_tanh(float x) {
  return tanhf(fmaxf(x, 0.0f));
}

// ---------------------------------------------------------------------------
// 1) nv1/nv2 = tanh(ALPHA * (S @ W^T)); one thread per (row, dout)
// ---------------------------------------------------------------------------
__global__ void prep_nv_kernel(const float* __restrict__ S,
                               const float* __restrict__ W1,
                               const float* __restrict__ W2,
                               float* __restrict__ nv1,
                               float* __restrict__ nv2,
                               int total_rows /* B*N */) {
  int idx = blockIdx.x * blockDim.x + threadIdx.x;     // over B*N*LD
  if (idx >= total_rows * LD) return;
  int rowid = idx >> 5;
  int dout  = idx & (LD - 1);
  const float* x  = S  + (long)rowid * LD;
  const float* w1 = W1 + (long)dout * LD;   // Linear: x @ W^T -> row dout of W
  const float* w2 = W2 + (long)dout * LD;
  float a1 = 0.f, a2 = 0.f;
#pragma unroll
  for (int k = 0; k < LD; ++k) {
    float xv = x[k];
    a1 = fmaf(xv, w1[k], a1);
    a2 = fmaf(xv, w2[k], a2);
  }
  nv1[(long)rowid * LD + dout] = tanhf(ALPHA_C * a1);
  nv2[(long)rowid * LD + dout] = tanhf(ALPHA_C * a2);
}

// ---------------------------------------------------------------------------
// 2) rank-1 vectors: s_st1 = S@w_st[:D], s_ts2 = S@w_ts[D:],
//                    s_st2 = Tn@w_st[D:], s_ts1 = Tn@w_ts[:D]
// ---------------------------------------------------------------------------
__global__ void prep_s_kernel(const float* __restrict__ S,
                              const float* __restrict__ Tn,
                              const float* __restrict__ w_st,
                              const float* __restrict__ w_ts,
                              float* __restrict__ s_st1, float* __restrict__ s_ts2,
                              float* __restrict__ s_st2, float* __restrict__ s_ts1,
                              int BN, int BT) {
  int idx = blockIdx.x * blockDim.x + threadIdx.x;
  if (idx < BN) {
    const float* x = S + (long)idx * LD;
    float a = 0.f, c = 0.f;
#pragma unroll
    for (int k = 0; k < LD; ++k) {
      a = fmaf(x[k], w_st[k], a);
      c = fmaf(x[k], w_ts[LD + k], c);
    }
    s_st1[idx] = a;
    s_ts2[idx] = c;
  } else if (idx < BN + BT) {
    int j = idx - BN;
    const float* x = Tn + (long)j * LD;
    float a = 0.f, c = 0.f;
#pragma unroll
    for (int k = 0; k < LD; ++k) {
      a = fmaf(x[k], w_st[LD + k], a);
      c = fmaf(x[k], w_ts[k], c);
    }
    s_st2[j] = a;
    s_ts1[j] = c;
  }
}

// ---------------------------------------------------------------------------
// 3) ss block: one wave per 16x16 output tile.
//    c1 = nv1[I]·nv2[J]^T (= m tile),  c2 = nv2[I]·nv1[J]^T (= m^T tile)
//    out = tanh(relu(tanh(c1 - c2)))
//    A layout (f32 16x4): lane l holds row l&15, pair K = 4*kk + 2*(l>>4).
//    C/D layout: VGPR v, lane l -> M = v + 8*(l>>4), N = l&15.
// ---------------------------------------------------------------------------
__global__ void ss_wmma_kernel(const float* __restrict__ nv1,
                               const float* __restrict__ nv2,
                               float* __restrict__ adj,
                               int B, int N, int M) {
  const int lane = threadIdx.x & 31;
  const int wave = threadIdx.x >> 5;
  const int tilesPerRow = N >> 4;                       // 128
  const long tilesPerBatch = (long)tilesPerRow * tilesPerRow;
  long tile = (long)blockIdx.x * (blockDim.x >> 5) + wave;
  if (tile >= (long)B * tilesPerBatch) return;
  int  b  = (int)(tile / tilesPerBatch);
  long r  = tile % tilesPerBatch;
  int  ti = (int)(r / tilesPerRow);
  int  tj = (int)(r % tilesPerRow);

  const int row = lane & 15;
  const int kof = (lane >> 4) << 1;                     // 0 or 2

  const float* a1 = nv1 + ((long)b * N + ti * 16 + row) * LD;   // A of C1
  const float* a2 = nv2 + ((long)b * N + ti * 16 + row) * LD;   // A of C2
  const float* b1 = nv2 + ((long)b * N + tj * 16 + row) * LD;   // B of C1 (col=row)
  const float* b2 = nv1 + ((long)b * N + tj * 16 + row) * LD;   // B of C2

  v8f c1 = {}; v8f c2 = {};
#pragma unroll
  for (int kk = 0; kk < LD / 4; ++kk) {
    const int k = kk * 4 + kof;
    v2f A1 = { a1[k], a1[k + 1] };
    v2f B1 = { b1[k], b1[k + 1] };
    v2f A2 = { a2[k], a2[k + 1] };
    v2f B2 = { b2[k], b2[k + 1] };
    c1 = __builtin_amdgcn_wmma_f32_16x16x4_f32(false, A1, false, B1,
                                               (short)0, c1, false, false);
    c2 = __builtin_amdgcn_wmma_f32_16x16x4_f32(false, A2, false, B2,
                                               (short)0, c2, false, false);
  }

  float* outb = adj + (long)b * M * M;
  const int col = tj * 16 + (lane & 15);
  const int mofs = (lane >> 4) << 3;                    // 0 or 8
#pragma unroll
  for (int v = 0; v < 8; ++v) {
    int gi = ti * 16 + v + mofs;
    float val = tanhf(c1[v] - c2[v]);                   // tanh(m - m^T)
    outb[(long)gi * M + col] = act_relu_tanh(val);      // tanh(relu(.))
  }
}

// ---------------------------------------------------------------------------
// 4) tt block: C = Tn[I]·Tn[J]^T, triu mask, tanh(relu(.))
// ---------------------------------------------------------------------------
__global__ void tt_wmma_kernel(const float* __restrict__ Tn,
                               float* __restrict__ adj,
                               int B, int N, int T, int M) {
  const int lane = threadIdx.x & 31;
  const int wave = threadIdx.x >> 5;
  const int tilesPerRow = T >> 4;                       // 18
  const long tilesPerBatch = (long)tilesPerRow * tilesPerRow;
  long tile = (long)blockIdx.x * (blockDim.x >> 5) + wave;
  if (tile >= (long)B * tilesPerBatch) return;
  int  b  = (int)(tile / tilesPerBatch);
  long r  = tile % tilesPerBatch;
  int  ti = (int)(r / tilesPerRow);
  int  tj = (int)(r % tilesPerRow);

  const int row = lane & 15;
  const int kof = (lane >> 4) << 1;

  const float* ap = Tn + ((long)b * T + ti * 16 + row) * LD;
  const float* bp = Tn + ((long)b * T + tj * 16 + row) * LD;

  v8f c = {};
#pragma unroll
  for (int kk = 0; kk < LD / 4; ++kk) {
    const int k = kk * 4 + kof;
    v2f A = { ap[k], ap[k + 1] };
    v2f Bv = { bp[k], bp[k + 1] };
    c = __builtin_amdgcn_wmma_f32_16x16x4_f32(false, A, false, Bv,
                                              (short)0, c, false, false);
  }

  float* outb = adj + (long)b * M * M;
  const int colL = tj * 16 + (lane & 15);               // tt-local col
  const int mofs = (lane >> 4) << 3;
#pragma unroll
  for (int v = 0; v < 8; ++v) {
    int rowL = ti * 16 + v + mofs;                      // tt-local row
    float val = (colL >= rowL) ? act_relu_tanh(c[v]) : 0.0f;  // triu
    outb[(long)(N + rowL) * M + (N + colL)] = val;
  }
}

// ---------------------------------------------------------------------------
// 5) st / ts blocks: one thread -> one st element + one ts element
// ---------------------------------------------------------------------------
__global__ void fill_st_ts_kernel(const float* __restrict__ s_st1,
                                  const float* __restrict__ s_st2,
                                  const float* __restrict__ s_ts1,
                                  const float* __restrict__ s_ts2,
                                  const float* __restrict__ b_st,
                                  const float* __restrict__ b_ts,
                                  float* __restrict__ adj,
                                  int B, int N, int T, int M) {
  long idx = (long)blockIdx.x * blockDim.x + threadIdx.x;   // over B*N*T
  long total = (long)B * N * T;
  if (idx >= total) return;
  int t = (int)(idx % T);
  long rr = idx / T;
  int n = (int)(rr % N);
  int b = (int)(rr / N);

  float stv = act_relu_tanh(s_st1[b * N + n] + s_st2[b * T + t] + b_st[0]);
  float tsv = act_relu_tanh(s_ts1[b * T + t] + s_ts2[b * N + n] + b_ts[0]);

  float* base = adj + (long)b * M * M;
  base[(long)n * M + (N + t)] = stv;          // adj[b, n, N+t]
  base[(long)(N + t) * M + n] = tsv;          // adj[b, N+t, n]
}

// ---------------------------------------------------------------------------
extern "C" void kernel_launch(void* const* d_in, const int* in_sizes, int n_in,
                              void* d_out, int out_size, void* d_ws, size_t ws_size,
                              hipStream_t stream) {
  (void)in_sizes; (void)n_in; (void)out_size; (void)ws_size;

  const int B = 4, N = 2048, T = 288, M = N + T;

  const float* S     = (const float*)d_in[0];   // [B,N,D]
  const float* Tn    = (const float*)d_in[1];   // [B,T,D]
  const float* W_ss1 = (const float*)d_in[2];   // [D,D]
  const float* W_ss2 = (const float*)d_in[3];   // [D,D]
  const float* w_st  = (const float*)d_in[4];   // [2D]
  const float* b_st  = (const float*)d_in[5];   // [1]
  const float* w_ts  = (const float*)d_in[6];   // [2D]
  const float* b_ts  = (const float*)d_in[7];   // [1]
  float* adj = (float*)d_out;                   // [B,M,M]

  // workspace layout (floats)
  float* ws    = (float*)d_ws;
  float* nv1   = ws;                       // B*N*LD
  float* nv2   = nv1 + (long)B * N * LD;   // B*N*LD
  float* s_st1 = nv2 + (long)B * N * LD;   // B*N
  float* s_ts2 = s_st1 + (long)B * N;      // B*N
  float* s_st2 = s_ts2 + (long)B * N;      // B*T
  float* s_ts1 = s_st2 + (long)B * T;      // B*T

  // 1) nv1/nv2
  {
    int threads = B * N * LD;
    prep_nv_kernel<<<(threads + 255) / 256, 256, 0, stream>>>(
        S, W_ss1, W_ss2, nv1, nv2, B * N);
  }
  // 2) rank-1 vectors
  {
    int threads = B * N + B * T;
    prep_s_kernel<<<(threads + 255) / 256, 256, 0, stream>>>(
        S, Tn, w_st, w_ts, s_st1, s_ts2, s_st2, s_ts1, B * N, B * T);
  }
  // 3) ss block: B*(N/16)^2 = 65536 tiles, 8 waves (tiles) per 256-thread block
  {
    long tiles = (long)B * (N / 16) * (N / 16);
    int blocks = (int)((tiles + 7) / 8);
    ss_wmma_kernel<<<blocks, 256, 0, stream>>>(nv1, nv2, adj, B, N, M);
  }
  // 4) tt block: B*(T/16)^2 = 1296 tiles
  {
    long tiles = (long)B * (T / 16) * (T / 16);
    int blocks = (int)((tiles + 7) / 8);
    tt_wmma_kernel<<<blocks, 256, 0, stream>>>(Tn, adj, B, N, T, M);
  }
  // 5) st / ts blocks
  {
    long threads = (long)B * N * T;
    int blocks = (int)((threads + 255) / 256);
    fill_st_ts_kernel<<<blocks, 256, 0, stream>>>(
        s_st1, s_st2, s_ts1, s_ts2, b_st, b_ts, adj, B, N, T, M);
  }
}